// WHModel_37512244363421
// MI455X (gfx1250) — compile-verified
//
#include <hip/hip_runtime.h>
#include <hip/hip_bf16.h>
#include <math.h>

// ---------------------------------------------------------------------------
// Wiener-Hammerstein model, chunked parallel-scan formulation for gfx1250.
//
//   LTI1 (scan) -> tanh MLP (pointwise) -> LTI2 (scan)
//
// Each LTI scan y_t = C(A x_{t-1} + B u_t) + D u_t  (x_0 = 0) is decomposed
// into 64 chunks of L=64 steps:
//   y_local   = U (64 chunks x 64) * Toeplitz(ir)      <- WMMA f32 GEMM
//   x_end     = U * S (64x8, S[j] = A^{L-1-j} B)       <- WMMA f32 GEMM
//   x_in[c+1] = A^L x_in[c] + x_end[c]                 <- 64-step 5x5 matvec
//   y         = y_local + Xin (64x8) * G^T (8x64)      <- WMMA f32 GEMM
// with ir[0] = C.B + D, ir[m] = C A^m B, G[t'] = C A^{t'+1}.
// The tanh MLP is fused into the correction pass of system 1.
// ---------------------------------------------------------------------------

typedef float v2f __attribute__((ext_vector_type(2)));
typedef float v8f __attribute__((ext_vector_type(8)));

#define BATCH    2048
#define TLEN     4096
#define LCHUNK   64
#define NCHUNK   64      // TLEN / LCHUNK
#define PRE_SZ   1120    // floats per (system,batch): ir[64] + S[64*8] + G[64*8] + AL[32]

// ---------------- K0: per-(system,batch) precompute --------------------------
__global__ void wh_precompute(const float* __restrict__ p, float* __restrict__ pre) {
    int idx = blockIdx.x * blockDim.x + threadIdx.x;   // 0 .. 2*BATCH-1
    if (idx >= 2 * BATCH) return;
    int s = idx >> 11;          // system 0/1
    int b = idx & (BATCH - 1);
    const float* pb = p + (size_t)b * 160 + (s ? 124 : 0);

    float A[5][5], Bv[5], Cv[5], Dv;
    #pragma unroll
    for (int i = 0; i < 5; ++i)
        #pragma unroll
        for (int j = 0; j < 5; ++j) A[i][j] = pb[i * 5 + j];
    #pragma unroll
    for (int i = 0; i < 5; ++i) { Bv[i] = pb[25 + i]; Cv[i] = pb[30 + i]; }
    Dv = pb[35];

    float* out = pre + (size_t)idx * PRE_SZ;
    float* ir  = out;              // [64]
    float* S   = out + 64;         // [64][8]   S[j][i] = (A^{63-j} B)_i
    float* G   = out + 64 + 512;   // [64][8]   G[m][i] = (C A^{m+1})_i
    float* AL  = out + 64 + 1024;  // [32]      A^64 row-major (25 used)

    // ir[0] = C.B + D ; S row 63 = A^0 B
    float acc0 = Dv;
    #pragma unroll
    for (int i = 0; i < 5; ++i) acc0 += Cv[i] * Bv[i];
    ir[0] = acc0;

    float w[5];
    #pragma unroll
    for (int i = 0; i < 5; ++i) w[i] = Bv[i];
    #pragma unroll
    for (int i = 0; i < 8; ++i) S[63 * 8 + i] = (i < 5) ? w[i] : 0.0f;

    // r = C A ; G row 0
    float r[5];
    #pragma unroll
    for (int j = 0; j < 5; ++j) {
        float t = 0.f;
        #pragma unroll
        for (int i = 0; i < 5; ++i) t += Cv[i] * A[i][j];
        r[j] = t;
    }
    #pragma unroll
    for (int i = 0; i < 8; ++i) G[i] = (i < 5) ? r[i] : 0.0f;

    for (int m = 1; m < LCHUNK; ++m) {
        float wn[5], rn[5];
        #pragma unroll
        for (int i = 0; i < 5; ++i) {
            float t = 0.f;
            #pragma unroll
            for (int j = 0; j < 5; ++j) t += A[i][j] * w[j];
            wn[i] = t;
        }
        #pragma unroll
        for (int j = 0; j < 5; ++j) {
            float t = 0.f;
            #pragma unroll
            for (int i = 0; i < 5; ++i) t += r[i] * A[i][j];
            rn[j] = t;
        }
        float irm = 0.f;
        #pragma unroll
        for (int i = 0; i < 5; ++i) { w[i] = wn[i]; r[i] = rn[i]; irm += Cv[i] * wn[i]; }
        ir[m] = irm;
        #pragma unroll
        for (int i = 0; i < 8; ++i) {
            S[(63 - m) * 8 + i] = (i < 5) ? w[i] : 0.0f;
            G[m * 8 + i]        = (i < 5) ? r[i] : 0.0f;
        }
    }

    // A^64 by repeated multiply
    float Ap[5][5], Tm[5][5];
    #pragma unroll
    for (int i = 0; i < 5; ++i)
        #pragma unroll
        for (int j = 0; j < 5; ++j) Ap[i][j] = A[i][j];
    for (int m = 1; m < LCHUNK; ++m) {
        #pragma unroll
        for (int i = 0; i < 5; ++i)
            #pragma unroll
            for (int j = 0; j < 5; ++j) {
                float t = 0.f;
                #pragma unroll
                for (int k = 0; k < 5; ++k) t += A[i][k] * Ap[k][j];
                Tm[i][j] = t;
            }
        #pragma unroll
        for (int i = 0; i < 5; ++i)
            #pragma unroll
            for (int j = 0; j < 5; ++j) Ap[i][j] = Tm[i][j];
    }
    #pragma unroll
    for (int i = 0; i < 32; ++i) AL[i] = 0.0f;
    #pragma unroll
    for (int i = 0; i < 5; ++i)
        #pragma unroll
        for (int j = 0; j < 5; ++j) AL[i * 5 + j] = Ap[i][j];
}

// ---------------- K1: chunk-local GEMMs (WMMA f32 16x16x4) -------------------
// One workgroup (4 waves) per batch.  Y(64x64) = U * Toeplitz(ir),
// Xend(64x8) = U * S.  Each wave owns a 16-chunk-row strip.
__global__ void __launch_bounds__(128)
wh_chunk_gemm(const float* __restrict__ u, const float* __restrict__ pre_sys,
              float* __restrict__ ybuf, float* __restrict__ xend) {
    __shared__ float Uld[64][65];
    __shared__ float Mld[64][65];
    __shared__ float Sld[64][8];
    __shared__ float irld[64];

    int b = blockIdx.x;
    int tid = threadIdx.x;
    const float* pre = pre_sys + (size_t)b * PRE_SZ;

    if (tid < 64) irld[tid] = pre[tid];
    for (int i = tid; i < 512; i += 128) Sld[i >> 3][i & 7] = pre[64 + i];
    for (int i = tid; i < 4096; i += 128) Uld[i >> 6][i & 63] = u[(size_t)b * TLEN + i];
    __syncthreads();
    for (int i = tid; i < 4096; i += 128) {
        int j = i >> 6, t = i & 63;
        Mld[j][t] = (t >= j) ? irld[t - j] : 0.0f;   // upper-tri Toeplitz
    }
    __syncthreads();

    int wave = tid >> 5, lane = tid & 31;
    int half = lane >> 4, lrow = lane & 15;
    int rowbase = wave * 16;

    // Y = U * M : 4 tiles of 16x16, K = 64 in steps of 4
    for (int n = 0; n < 4; ++n) {
        v8f acc = {};
        #pragma unroll
        for (int k = 0; k < 64; k += 4) {
            v2f a, bb;
            a.x  = Uld[rowbase + lrow][k + 2 * half];
            a.y  = Uld[rowbase + lrow][k + 2 * half + 1];
            bb.x = Mld[k + 2 * half][n * 16 + lrow];
            bb.y = Mld[k + 2 * half + 1][n * 16 + lrow];
            acc = __builtin_amdgcn_wmma_f32_16x16x4_f32(
                false, a, false, bb, (short)0, acc, false, false);
        }
        #pragma unroll
        for (int v = 0; v < 8; ++v) {
            int c = rowbase + v + 8 * half;
            int t = n * 16 + lrow;
            ybuf[(size_t)b * TLEN + c * 64 + t] = acc[v];
        }
    }

    // Xend = U * S : one 16x16 tile (cols 8..15 zero-padded)
    {
        v8f acc = {};
        #pragma unroll
        for (int k = 0; k < 64; k += 4) {
            v2f a, bb;
            a.x  = Uld[rowbase + lrow][k + 2 * half];
            a.y  = Uld[rowbase + lrow][k + 2 * half + 1];
            bb.x = (lrow < 8) ? Sld[k + 2 * half][lrow] : 0.0f;
            bb.y = (lrow < 8) ? Sld[k + 2 * half + 1][lrow] : 0.0f;
            acc = __builtin_amdgcn_wmma_f32_16x16x4_f32(
                false, a, false, bb, (short)0, acc, false, false);
        }
        if (lrow < 8) {
            #pragma unroll
            for (int v = 0; v < 8; ++v) {
                int c = rowbase + v + 8 * half;
                xend[((size_t)b * NCHUNK + c) * 8 + lrow] = acc[v];
            }
        }
    }
}

// ---------------- K2: sequential inter-chunk state propagation ---------------
__global__ void wh_propagate(const float* __restrict__ pre_sys,
                             const float* __restrict__ xend,
                             float* __restrict__ xin) {
    int b = blockIdx.x * blockDim.x + threadIdx.x;
    if (b >= BATCH) return;
    const float* AL = pre_sys + (size_t)b * PRE_SZ + 64 + 1024;
    float M[25];
    #pragma unroll
    for (int i = 0; i < 25; ++i) M[i] = AL[i];
    float x[5] = {0.f, 0.f, 0.f, 0.f, 0.f};
    for (int c = 0; c < NCHUNK; ++c) {
        size_t base = ((size_t)b * NCHUNK + c) * 8;
        #pragma unroll
        for (int i = 0; i < 5; ++i) xin[base + i] = x[i];
        xin[base + 5] = 0.f; xin[base + 6] = 0.f; xin[base + 7] = 0.f;
        float xn[5];
        #pragma unroll
        for (int i = 0; i < 5; ++i) {
            float t = xend[base + i];
            #pragma unroll
            for (int j = 0; j < 5; ++j) t += M[i * 5 + j] * x[j];
            xn[i] = t;
        }
        #pragma unroll
        for (int i = 0; i < 5; ++i) x[i] = xn[i];
    }
}

// ---------------- K3: correction GEMM (+ fused tanh MLP) ---------------------
// Ycorr(64x64) = Xin(64x8) * G^T(8x64); y = y_local + Ycorr; optional MLP.
__global__ void __launch_bounds__(128)
wh_correct(const float* __restrict__ pre_sys, const float* __restrict__ xin,
           const float* __restrict__ ybuf, const float* __restrict__ p,
           float* __restrict__ outbuf, int apply_mlp) {
    __shared__ float Gt[8][65];    // Gt[i][t'] = (C A^{t'+1})_i
    __shared__ float Xld[64][8];
    __shared__ float mlp[96];      // w1[29], b1[29], w2[29], b2

    int b = blockIdx.x;
    int tid = threadIdx.x;
    const float* pre = pre_sys + (size_t)b * PRE_SZ;

    for (int i = tid; i < 512; i += 128) {
        int t = i >> 3, st = i & 7;
        Gt[st][t] = pre[64 + 512 + i];                       // G[t][st]
        Xld[t][st] = xin[((size_t)b * NCHUNK + t) * 8 + st];
    }
    if (tid < 88) mlp[tid] = p[(size_t)b * 160 + 36 + tid];
    __syncthreads();

    int wave = tid >> 5, lane = tid & 31;
    int half = lane >> 4, lrow = lane & 15;
    int rowbase = wave * 16;

    for (int n = 0; n < 4; ++n) {
        v8f acc = {};
        #pragma unroll
        for (int k = 0; k < 8; k += 4) {
            v2f a, bb;
            a.x  = Xld[rowbase + lrow][k + 2 * half];
            a.y  = Xld[rowbase + lrow][k + 2 * half + 1];
            bb.x = Gt[k + 2 * half][n * 16 + lrow];
            bb.y = Gt[k + 2 * half + 1][n * 16 + lrow];
            acc = __builtin_amdgcn_wmma_f32_16x16x4_f32(
                false, a, false, bb, (short)0, acc, false, false);
        }
        #pragma unroll
        for (int v = 0; v < 8; ++v) {
            int c = rowbase + v + 8 * half;
            int t = n * 16 + lrow;
            size_t gidx = (size_t)b * TLEN + c * 64 + t;
            float y = ybuf[gidx] + acc[v];
            if (apply_mlp) {
                float sacc = mlp[87];                         // b2
                #pragma unroll 1
                for (int h = 0; h < 29; ++h)
                    sacc += mlp[58 + h] * tanhf(y * mlp[h] + mlp[29 + h]);
                outbuf[gidx] = sacc;
            } else {
                outbuf[gidx] = y;
            }
        }
    }
}

// ---------------- host launcher ---------------------------------------------
extern "C" void kernel_launch(void* const* d_in, const int* in_sizes, int n_in,
                              void* d_out, int out_size, void* d_ws, size_t ws_size,
                              hipStream_t stream) {
    const float* p = (const float*)d_in[0];   // (2048, 160)
    const float* u = (const float*)d_in[1];   // (2048, 4096, 1)
    float* out = (float*)d_out;               // (2048, 4096, 1)
    float* ws = (float*)d_ws;

    // workspace layout (floats)
    const size_t PRE_TOT = (size_t)2 * BATCH * PRE_SZ;          // 4,587,520
    float* pre  = ws;
    float* ybuf = ws + PRE_TOT;                                 // B*T
    float* vbuf = ybuf + (size_t)BATCH * TLEN;                  // B*T
    float* xend = vbuf + (size_t)BATCH * TLEN;                  // B*64*8
    float* xin  = xend + (size_t)BATCH * NCHUNK * 8;            // B*64*8

    float* pre0 = pre;
    float* pre1 = pre + (size_t)BATCH * PRE_SZ;

    wh_precompute<<<(2 * BATCH + 127) / 128, 128, 0, stream>>>(p, pre);

    // System 1: u -> y1, MLP fused -> v
    wh_chunk_gemm<<<BATCH, 128, 0, stream>>>(u, pre0, ybuf, xend);
    wh_propagate<<<BATCH / 128, 128, 0, stream>>>(pre0, xend, xin);
    wh_correct<<<BATCH, 128, 0, stream>>>(pre0, xin, ybuf, p, vbuf, 1);

    // System 2: v -> y2 -> out
    wh_chunk_gemm<<<BATCH, 128, 0, stream>>>(vbuf, pre1, ybuf, xend);
    wh_propagate<<<BATCH / 128, 128, 0, stream>>>(pre1, xend, xin);
    wh_correct<<<BATCH, 128, 0, stream>>>(pre1, xin, ybuf, p, out, 0);
}